// HarmonicGenerator_60765197303837
// MI455X (gfx1250) — compile-verified
//
#include <hip/hip_runtime.h>
#include <math.h>

// ---------------- problem constants (from reference) ----------------
#define NB     32            // batch
#define CCH    256           // channels (K of the GEMM)
#define LF     250           // frames per batch
#define SEG    960           // upsample factor
#define LW     (LF * SEG)    // 240000 samples per batch
#define NH     8             // harmonics
#define NROWS  (NB * LF)     // 8000 GEMM rows
#define NTILES (NROWS / 16)  // 500 16-row WMMA tiles (divides exactly)
#define NCHUNK (NB * LF)     // 8000 chunks of 960 samples
#define SPL    (SEG / 32)    // 30 samples per lane (wave32)
#define INV_SR (1.0f / 48000.0f)
#define BASEF  220.0f
#define TWO_PI 6.28318530717958647692f

typedef __attribute__((ext_vector_type(2))) float v2f;
typedef __attribute__((ext_vector_type(8))) float v8f;

// Linear interp of a 250-long frame array at upsampled sample s
// (matches torch interpolate(mode='linear', align_corners=False)).
__device__ __forceinline__ float interp_frames(const float* __restrict__ base, int s) {
    float pos = ((float)s + 0.5f) * (1.0f / (float)SEG) - 0.5f;
    pos = fminf(fmaxf(pos, 0.0f), (float)(LF - 1));
    int   i0 = (int)floorf(pos);
    int   i1 = (i0 + 1 < LF) ? (i0 + 1) : (LF - 1);
    float w  = pos - (float)i0;
    return base[i0] * (1.0f - w) + base[i1] * w;
}

// ---------------------------------------------------------------
// Kernel 1: frame-rate GEMM via V_WMMA_F32_16X16X4_F32.
// One wave = one 16(M) x 16(N) tile; M = batch*250 + frame, N = output
// (o==0 -> octave, o in 1..8 -> harmonic mags, o>8 -> zero pad).
// Emits f0[row] = 220*2^octave and magMean[row] = mean_h exp(min(.,6)).
// ---------------------------------------------------------------
__global__ void frame_gemm_wmma(const float* __restrict__ x,
                                const float* __restrict__ w_mag,
                                const float* __restrict__ b_mag,
                                const float* __restrict__ w_oct,
                                const float* __restrict__ b_oct,
                                float* __restrict__ f0f,
                                float* __restrict__ mg) {
    int wave = blockIdx.x * (blockDim.x >> 5) + (threadIdx.x >> 5);
    if (wave >= NTILES) return;                 // wave-uniform: EXEC stays all-1s
    int lane = threadIdx.x & 31;
    int m0   = wave * 16;

    // A-matrix addressing: lanes 0-15 -> M rows, VGPR pair holds K/K+1,
    // lane[4] selects K+0/K+1 vs K+2/K+3 (ISA 16x4 f32 A layout).
    int aM = m0 + (lane & 15);
    int an = aM / LF;
    int al = aM - an * LF;
    const float* xa  = x + (size_t)an * (CCH * LF) + al;   // stride LF over channels
    int kSel = (lane >> 4) << 1;                           // 0 or 2

    // B-matrix: lanes hold N (output o), same K-pair split as A.
    int o = lane & 15;
    const float* wp;
    float wscale = 1.0f;
    if (o == 0)            { wp = w_oct; }
    else if (o <= NH)      { wp = w_mag + (o - 1) * CCH; }
    else                   { wp = w_oct; wscale = 0.0f; }  // pad columns -> 0

    v8f acc = {};
    #pragma unroll 4
    for (int k = 0; k < CCH; k += 4) {
        v2f a, b;
        a.x = xa[(k + kSel)     * LF];   // 16 lanes -> consecutive l: coalesced
        a.y = xa[(k + kSel + 1) * LF];
        b.x = wp[k + kSel]     * wscale;
        b.y = wp[k + kSel + 1] * wscale;
        acc = __builtin_amdgcn_wmma_f32_16x16x4_f32(
            /*neg_a=*/false, a, /*neg_b=*/false, b,
            /*c_mod=*/(short)0, acc, /*reuse_a=*/false, /*reuse_b=*/false);
    }

    float bias = (o == 0) ? b_oct[0] : ((o <= NH) ? b_mag[o - 1] : 0.0f);

    // C/D layout: VGPR r -> M = m0 + r (lanes 0-15) or m0 + r + 8 (lanes 16-31).
    #pragma unroll
    for (int r = 0; r < 8; ++r) {
        float v    = acc[r] + bias;
        float magv = expf(fminf(v, 6.0f));
        // Sum mags over lanes o=1..8 inside each 16-lane half (butterfly).
        float contrib = (o >= 1 && o <= NH) ? magv : 0.0f;
        contrib += __shfl_xor(contrib, 1, 32);
        contrib += __shfl_xor(contrib, 2, 32);
        contrib += __shfl_xor(contrib, 4, 32);
        contrib += __shfl_xor(contrib, 8, 32);
        if (o == 0) {                       // lanes 0 and 16 own rows r and r+8
            int m  = m0 + r + ((lane >> 4) << 3);
            f0f[m] = BASEF * exp2f(v);      // v is the octave output on o==0
            mg[m]  = contrib * (1.0f / (float)NH);
        }
    }
}

// ---------------------------------------------------------------
// Kernel 2: per-chunk (960-sample) formant sums. One wave per chunk,
// 30 samples per lane, wave32 butterfly reduction.
// ---------------------------------------------------------------
__global__ void chunk_sums(const float* __restrict__ f0f,
                           float* __restrict__ chunkSum) {
    int wave = blockIdx.x * (blockDim.x >> 5) + (threadIdx.x >> 5);
    if (wave >= NCHUNK) return;
    int lane = threadIdx.x & 31;
    int n  = wave / LF;
    int cc = wave - n * LF;
    const float* fbase = f0f + n * LF;
    int s0 = cc * SEG + lane * SPL;

    float sum = 0.0f;
    #pragma unroll
    for (int i = 0; i < SPL; ++i) sum += interp_frames(fbase, s0 + i);

    sum += __shfl_xor(sum, 1, 32);
    sum += __shfl_xor(sum, 2, 32);
    sum += __shfl_xor(sum, 4, 32);
    sum += __shfl_xor(sum, 8, 32);
    sum += __shfl_xor(sum, 16, 32);
    if (lane == 0) chunkSum[wave] = sum;
}

// ---------------------------------------------------------------
// Kernel 3: exclusive prefix over the 250 chunk sums of each batch.
// One wave32 per batch; 8 strided wave scans with a carry.
// ---------------------------------------------------------------
__global__ void chunk_scan(const float* __restrict__ chunkSum,
                           float* __restrict__ chunkPrefix) {
    int n    = blockIdx.x;
    int lane = threadIdx.x;     // blockDim.x == 32
    float carry = 0.0f;
    for (int base = 0; base < LF; base += 32) {
        int   idx = base + lane;
        float v   = (idx < LF) ? chunkSum[n * LF + idx] : 0.0f;
        float incl = v;
        #pragma unroll
        for (int d = 1; d < 32; d <<= 1) {
            float y = __shfl_up(incl, d, 32);
            if (lane >= d) incl += y;
        }
        if (idx < LF) chunkPrefix[n * LF + idx] = carry + (incl - v);  // exclusive
        carry += __shfl(incl, 31, 32);
    }
}

// ---------------------------------------------------------------
// Kernel 4: final synthesis. One wave per chunk. Lane-local formant
// sums -> wave32 exclusive scan -> inclusive running phase per sample,
// then stream sin(2*pi*t) * magMean to the output (the only 30 MB write).
// ---------------------------------------------------------------
__global__ void synth(const float* __restrict__ f0f,
                      const float* __restrict__ mg,
                      const float* __restrict__ chunkPrefix,
                      const float* __restrict__ phi,
                      float* __restrict__ out) {
    int wave = blockIdx.x * (blockDim.x >> 5) + (threadIdx.x >> 5);
    if (wave >= NCHUNK) return;
    int lane = threadIdx.x & 31;
    int n  = wave / LF;
    int cc = wave - n * LF;
    const float* fbase = f0f + n * LF;
    const float* mbase = mg  + n * LF;
    int s0 = cc * SEG + lane * SPL;

    float fv[SPL];
    float localSum = 0.0f;
    #pragma unroll
    for (int i = 0; i < SPL; ++i) {
        fv[i] = interp_frames(fbase, s0 + i);
        localSum += fv[i];
    }

    float incl = localSum;
    #pragma unroll
    for (int d = 1; d < 32; d <<= 1) {
        float y = __shfl_up(incl, d, 32);
        if (lane >= d) incl += y;
    }
    float acc = chunkPrefix[wave] + (incl - localSum);   // exclusive up to my samples
    float ph  = phi[n];
    float* op = out + (size_t)n * LW;

    #pragma unroll
    for (int i = 0; i < SPL; ++i) {
        acc += fv[i];                                    // inclusive cumsum
        float t = ph + acc * INV_SR;
        float m = interp_frames(mbase, s0 + i);
        op[s0 + i] = __sinf(TWO_PI * t) * m;
    }
}

// ---------------------------------------------------------------
extern "C" void kernel_launch(void* const* d_in, const int* in_sizes, int n_in,
                              void* d_out, int out_size, void* d_ws, size_t ws_size,
                              hipStream_t stream) {
    const float* x     = (const float*)d_in[0];   // (32,256,250)
    const float* phi   = (const float*)d_in[1];   // (32,1,1)
    const float* w_mag = (const float*)d_in[2];   // (8,256,1)
    const float* b_mag = (const float*)d_in[3];   // (8,)
    const float* w_oct = (const float*)d_in[4];   // (1,256,1)
    const float* b_oct = (const float*)d_in[5];   // (1,)
    float* out = (float*)d_out;                   // (32,1,240000)

    float* f0f  = (float*)d_ws;        // 8000: f0 per (n,frame)
    float* mg   = f0f  + NROWS;        // 8000: mean mag per (n,frame)
    float* csum = mg   + NROWS;        // 8000: per-chunk formant sums
    float* cpre = csum + NCHUNK;       // 8000: exclusive chunk prefixes

    dim3 blk(256);  // 8 wave32s per block
    frame_gemm_wmma<<<dim3((NTILES + 7) / 8), blk, 0, stream>>>(
        x, w_mag, b_mag, w_oct, b_oct, f0f, mg);
    chunk_sums<<<dim3((NCHUNK + 7) / 8), blk, 0, stream>>>(f0f, csum);
    chunk_scan<<<dim3(NB), dim3(32), 0, stream>>>(csum, cpre);
    synth<<<dim3((NCHUNK + 7) / 8), blk, 0, stream>>>(f0f, mg, cpre, phi, out);
}